// Renderer_63247688400982
// MI455X (gfx1250) — compile-verified
//
#include <hip/hip_runtime.h>
#include <math.h>

typedef float v2f __attribute__((ext_vector_type(2)));
typedef float v8f __attribute__((ext_vector_type(8)));

// Plane-fit normal from 3x3 neighborhood sums (matches estimate_normals()).
__device__ __forceinline__ void plane_normal(
    float Sx, float Sy, float Sz, float Sxx, float Sxy, float Syy,
    float Sxz, float Syz, float& nx, float& ny, float& nz)
{
    const float N = 9.0f;
    float det = Sxx*(Syy*N - Sy*Sy) - Sxy*(Sxy*N - Sy*Sx) + Sx*(Sxy*Sy - Syy*Sx);
    det = (fabsf(det) < 1e-6f) ? 1e-6f : det;
    float da = Sxz*(Syy*N - Sy*Sy) - Sxy*(Syz*N - Sy*Sz) + Sx*(Syz*Sy - Syy*Sz);
    float db = Sxx*(Syz*N - Sz*Sy) - Sxz*(Sxy*N - Sy*Sx) + Sx*(Sxy*Sz - Syz*Sx);
    float a = da / det;
    float b = db / det;
    float tx = -a, ty = -b, tz = 1.0f;
    float r = 1.0f / (sqrtf(tx*tx + ty*ty + tz*tz) + 1e-8f);
    nx = tx*r; ny = ty*r; nz = tz*r;
}

// One block (256 threads = 8 wave32 waves) per outer pixel.
// Thread t = inner pixel (i1 = t/16, j1 = t%16) of the 16x16 fisheye grid.
__global__ __launch_bounds__(256) void renderer_kernel(
    const float* __restrict__ view, float* __restrict__ out)
{
    __shared__ float nb[9][3];                 // wc0 3x3 neighborhood
    __shared__ float wxs[256], wys[256], wzs[256]; // wc1 tile
    __shared__ float red[3][256];              // per-channel contributions

    const int tid = threadIdx.x;
    const int pix = blockIdx.x;
    const int i0  = pix >> 7;
    const int j0  = pix & 127;

    const float v0 = view[0], v1 = view[1], v2 = view[2];
    const float v3 = view[3], v4 = view[4], v5 = view[5];
    const float s0 = v0 + v1 + v2 + v3 + v4 + v5;

    // ---- level 0: wc0 at the 3x3 neighborhood (threads 0..8) ----
    if (tid < 9) {
        int di = tid / 3 - 1, dj = tid % 3 - 1;
        int ii = min(max(i0 + di, 0), 127);
        int jj = min(max(j0 + dj, 0), 127);
        // w = h = tan(15deg)*2*0.1 ; halfw = w/2 ; step = w/127
        const float halfw = 0.026794919243112274f;
        const float step  = 0.00042196723217499573f;
        float X = -halfw + (float)ii * step;
        float Y = -halfw + (float)jj * step;
        float nrm = sqrtf(X*X + Y*Y + 0.01f);      // focal^2 = 0.01
        float inv = 1.0f / nrm;
        // sin(acos(f/n))*cos(atan2(X,Y)) = Y/n ; *sin(...) = X/n ; cos(theta)=f/n
        float ac = Y * inv;      // coefficient of p = (-cdz, 0, cdx)
        float bc = X * inv;      // coefficient of up = (0,1,0)
        float zc = 0.1f * inv;   // coefficient of camera_dir
        float dx = -ac*v5 + zc*v3;
        float dy =  bc    + zc*v4;
        float dz =  ac*v3 + zc*v5;
        float g  = (2.0f*(float)ii + 3.0f*(float)jj) * (1.0f/127.0f);
        float depth = 2.0f + 0.5f * sinf(0.3f*s0 + 4.0f*g);
        nb[tid][0] = v0 + depth*dx;
        nb[tid][1] = v1 + depth*dy;
        nb[tid][2] = v2 + depth*dz;
    }
    __syncthreads();

    // ---- n0 (redundant per-thread; trivial cost) ----
    float Sx=0.f,Sy=0.f,Sz=0.f,Sxx=0.f,Sxy=0.f,Syy=0.f,Sxz=0.f,Syz=0.f;
    #pragma unroll
    for (int t = 0; t < 9; ++t) {
        float x = nb[t][0], y = nb[t][1], z = nb[t][2];
        Sx+=x; Sy+=y; Sz+=z; Sxx+=x*x; Sxy+=x*y; Syy+=y*y; Sxz+=x*z; Syz+=y*z;
    }
    float n0x, n0y, n0z;
    plane_normal(Sx,Sy,Sz,Sxx,Sxy,Syy,Sxz,Syz, n0x,n0y,n0z);

    const float px0 = nb[4][0], py0 = nb[4][1], pz0 = nb[4][2];
    const float s1  = px0 + py0 + pz0 + n0x + n0y + n0z;

    // ---- level 1: wc1 for this thread's inner pixel ----
    const int i1 = tid >> 4;
    const int j1 = tid & 15;
    const float phi1   = (float)i1 * 0.4188790204786391f;   // 2*pi/15
    const float theta1 = (float)j1 * 0.10471975511965977f;  // (pi/2)/15
    float sph = sinf(phi1),   cph = cosf(phi1);
    float sth = sinf(theta1), cth = cosf(theta1);
    float xr = sth*cph, yr = sth*sph, zr = cth;
    // p1 = cross(n0, up) = (-n0z, 0, n0x)
    float d1x = -xr*n0z + zr*n0x;
    float d1y =  yr     + zr*n0y;
    float d1z =  xr*n0x + zr*n0z;
    float g1 = (2.0f*(float)i1 + 3.0f*(float)j1) * (1.0f/15.0f);
    float depth1 = 2.0f + 0.5f * sinf(0.3f*s1 + 4.0f*g1);
    float wcx = px0 + depth1*d1x;
    float wcy = py0 + depth1*d1y;
    float wcz = pz0 + depth1*d1z;
    wxs[tid] = wcx; wys[tid] = wcy; wzs[tid] = wcz;
    __syncthreads();

    // ---- n1 from 16x16 LDS tile (edge-clamped 3x3) ----
    Sx=Sy=Sz=Sxx=Sxy=Syy=Sxz=Syz=0.f;
    #pragma unroll
    for (int di = -1; di <= 1; ++di) {
        int ii = min(max(i1 + di, 0), 15);
        #pragma unroll
        for (int dj = -1; dj <= 1; ++dj) {
            int jj  = min(max(j1 + dj, 0), 15);
            int idx = (ii << 4) + jj;
            float x = wxs[idx], y = wys[idx], z = wzs[idx];
            Sx+=x; Sy+=y; Sz+=z; Sxx+=x*x; Sxy+=x*y; Syy+=y*y; Sxz+=x*z; Syz+=y*z;
        }
    }
    float n1x, n1y, n1z;
    plane_normal(Sx,Sy,Sz,Sxx,Sxy,Syy,Sxz,Syz, n1x,n1y,n1z);
    float s2 = wcx + wcy + wcz + n1x + n1y + n1z;

    // ---- level 2 irradiance: theta2 = {0, fp32(pi/2)} ----
    // cos(0)*sin(0) == exactly 0 -> those terms vanish bit-exactly.
    // cosf(fp32(pi/2))*sinf(fp32(pi/2)) = -4.37113883e-8 * 1.0
    const float w2_1 = -4.37113883e-08f;
    float base = 0.2f * s2;
    float w1 = cth * sth;  // cos(theta1)*sin(theta1)
    #pragma unroll
    for (int c = 0; c < 3; ++c) {
        float m = (float)(c + 1);
        float acc2 = 0.0f;
        acc2 += 0.5f*(1.0f + cosf(base + 3.0f*m)) * w2_1;  // i2=0, j2=1
        acc2 += 0.5f*(1.0f + cosf(base + 5.0f*m)) * w2_1;  // i2=1, j2=1
        float out1c = 0.5f * 0.7853981633974483f * acc2;   // albedo0 * pi/4
        red[c][tid] = out1c * w1;
    }
    __syncthreads();

    // ---- deterministic 256->1 reduction per channel via V_WMMA_F32_16X16X4_F32 ----
    // A (16x4 f32 chunks) holds the contribution matrix, B = ones (layout-free),
    // C chained over 4 chunks: D[r,n] = sum_K A[r,K]. Lane-sum of the 8 D VGPRs
    // gives half-sums; shfl_xor(16) combines halves. Wave 0 only (EXEC all 1s).
    if (tid < 32) {
        const int m  = tid & 15;
        const int hi = tid >> 4;
        v2f bones; bones.x = 1.0f; bones.y = 1.0f;
        float res[3];
        #pragma unroll
        for (int c = 0; c < 3; ++c) {
            v8f acc = {0.f,0.f,0.f,0.f,0.f,0.f,0.f,0.f};
            #pragma unroll
            for (int kk = 0; kk < 4; ++kk) {
                v2f a;
                a.x = red[c][(4*kk + 2*hi + 0)*16 + m];
                a.y = red[c][(4*kk + 2*hi + 1)*16 + m];
                acc = __builtin_amdgcn_wmma_f32_16x16x4_f32(
                        false, a, false, bones, (short)0, acc, false, false);
            }
            float p = acc[0]+acc[1]+acc[2]+acc[3]+acc[4]+acc[5]+acc[6]+acc[7];
            p += __shfl_xor(p, 16, 32);
            res[c] = 0.5f * 0.012271846303085129f * p;  // albedo0 * pi/256
        }
        if (tid == 0) {
            float* o = out + pix * 3;
            o[0] = res[0]; o[1] = res[1]; o[2] = res[2];
        }
    }
}

extern "C" void kernel_launch(void* const* d_in, const int* in_sizes, int n_in,
                              void* d_out, int out_size, void* d_ws, size_t ws_size,
                              hipStream_t stream) {
    (void)in_sizes; (void)n_in; (void)d_ws; (void)ws_size; (void)out_size;
    // d_in[0] = z (1,256)  -- unused by the reference
    // d_in[1] = view (1,6)
    const float* view = (const float*)d_in[1];
    float* out = (float*)d_out;  // (1,128,128,3) f32
    hipLaunchKernelGGL(renderer_kernel, dim3(128 * 128), dim3(256), 0, stream,
                       view, out);
}